// Estimator_75179107549558
// MI455X (gfx1250) — compile-verified
//
#include <hip/hip_runtime.h>
#include <stdint.h>

// Problem geometry (fixed by the reference).
constexpr int kH = 224;
constexpr int kW = 398;
constexpr int kPlanes = 256 * 2;        // B * C depthwise planes

// Tiling: 4 x 7 tiles per plane (covers 400 x 224), minimal edge waste.
constexpr int TW = 100;                 // output tile width
constexpr int TH = 32;                  // output tile height
constexpr int PAD = 3;                  // blur halo (2) + maxpool halo (1)
constexpr int IN_W = TW + 2 * PAD;      // 106
constexpr int IN_H = TH + 2 * PAD;      // 38
constexpr int VB_W = IN_W;              // 106 (vertical-blur tile keeps x halo)
constexpr int VB_H = TH + 2;            // 34 (blur rows y0-1 .. y0+TH)
constexpr int B_W  = TW + 2;            // 102 (blur cols x0-1 .. x0+TW)
constexpr int B_H  = TH + 2;            // 34
constexpr int NTHREADS = 256;           // 8 wave32

// Gaussian(5, sigma=2) weights, normalized (symmetric: w0=w4, w1=w3).
constexpr float GW0 = 0.15246914f;
constexpr float GW1 = 0.22184130f;
constexpr float GW2 = 0.25137913f;
constexpr float THR = 0.3f;

__device__ __forceinline__ int reflect_idx(int i, int n) {
    // jnp.pad mode="reflect" (mirror excluding edge); |offset| <= 3 here, n >= 4.
    i = (i < 0) ? -i : i;
    i = (i >= n) ? (2 * n - 2 - i) : i;
    return i;
}

__global__ __launch_bounds__(NTHREADS)
void peaks_kernel(const float* __restrict__ in, float* __restrict__ out) {
    const int tid = threadIdx.x;
    const int x0 = blockIdx.x * TW;
    const int y0 = blockIdx.y * TH;
    const size_t plane_off = (size_t)blockIdx.z * (size_t)(kH * kW);
    const float* __restrict__ src = in + plane_off;
    float* __restrict__ dst = out + plane_off;

    __shared__ float sIn[IN_H * IN_W];
    __shared__ float sVB[VB_H * VB_W];
    __shared__ float sB[B_H * B_W];

    // ---- Stage 1: async global->LDS load of input tile + reflect halo ----
    for (int i = tid; i < IN_H * IN_W; i += NTHREADS) {
        const int ly = i / IN_W;
        const int lx = i - ly * IN_W;
        const int gy = reflect_idx(y0 + ly - PAD, kH);
        const int gx = reflect_idx(x0 + lx - PAD, kW);
        const uint64_t gaddr = (uint64_t)(uintptr_t)(src + gy * kW + gx);
        // Low 32 bits of the flat pointer to LDS == byte offset in group segment.
        const uint32_t laddr = (uint32_t)(uintptr_t)(&sIn[i]);
        asm volatile("global_load_async_to_lds_b32 %0, %1, off"
                     :: "v"(laddr), "v"(gaddr)
                     : "memory");
    }
    asm volatile("s_wait_asynccnt 0" ::: "memory");
    __syncthreads();

    // ---- Stage 2: vertical 5-tap blur (rows y0-1 .. y0+TH, keep x halo) ----
    for (int i = tid; i < VB_H * VB_W; i += NTHREADS) {
        const int ly = i / VB_W;
        const int lx = i - ly * VB_W;
        const float* c = &sIn[ly * IN_W + lx];  // taps at rows ly..ly+4
        sVB[i] = GW0 * (c[0]        + c[4 * IN_W])
               + GW1 * (c[IN_W]     + c[3 * IN_W])
               + GW2 *  c[2 * IN_W];
    }
    __syncthreads();

    // ---- Stage 3: horizontal 5-tap blur; -inf outside the image ----
    for (int i = tid; i < B_H * B_W; i += NTHREADS) {
        const int ly = i / B_W;
        const int lx = i - ly * B_W;
        const float* c = &sVB[ly * VB_W + lx];  // taps at cols lx..lx+4
        const float b = GW0 * (c[0] + c[4])
                      + GW1 * (c[1] + c[3])
                      + GW2 *  c[2];
        const int gy = y0 - 1 + ly;
        const int gx = x0 - 1 + lx;
        const bool valid = (gy >= 0) & (gy < kH) & (gx >= 0) & (gx < kW);
        sB[i] = valid ? b : -__builtin_huge_valf();
    }
    __syncthreads();

    // ---- Stage 4: 3x3 max, exact-equality NMS, threshold, NT store ----
    for (int i = tid; i < TH * TW; i += NTHREADS) {
        const int ly = i / TW;
        const int lx = i - ly * TW;
        const int gy = y0 + ly;
        const int gx = x0 + lx;
        if (gy < kH && gx < kW) {
            const float* c = &sB[ly * B_W + lx];
            const float b = c[B_W + 1];  // center (blur at gy,gx)
            float m = c[0];
            m = fmaxf(m, c[1]);
            m = fmaxf(m, c[2]);
            m = fmaxf(m, c[B_W]);
            m = fmaxf(m, b);
            m = fmaxf(m, c[B_W + 2]);
            m = fmaxf(m, c[2 * B_W]);
            m = fmaxf(m, c[2 * B_W + 1]);
            m = fmaxf(m, c[2 * B_W + 2]);
            const float r = ((b == m) && (b > THR)) ? b : 0.0f;
            // Output is write-once / never re-read: stream past L2 (NT hint)
            // so it doesn't evict input halo lines that neighbor tiles re-read.
            __builtin_nontemporal_store(r, &dst[gy * kW + gx]);
        }
    }
}

extern "C" void kernel_launch(void* const* d_in, const int* in_sizes, int n_in,
                              void* d_out, int out_size, void* d_ws, size_t ws_size,
                              hipStream_t stream) {
    const float* in = (const float*)d_in[0];
    float* out = (float*)d_out;
    (void)in_sizes; (void)n_in; (void)out_size; (void)d_ws; (void)ws_size;

    dim3 grid((kW + TW - 1) / TW,   // 4
              (kH + TH - 1) / TH,   // 7
              kPlanes);             // 512
    peaks_kernel<<<grid, NTHREADS, 0, stream>>>(in, out);
}